// DynamicConnectivity_42992622633708
// MI455X (gfx1250) — compile-verified
//
#include <hip/hip_runtime.h>
#include <hip/hip_bf16.h>

#define NB   64      // batch
#define NPRE 2048
#define NPOST 2048
#define NHIST 50
#define FORM_RATE 0.001f
#define PRUNE_THR 0.01f
#define MAX_SYN   100.0f
#define CORR_THR  0.1f

typedef __attribute__((ext_vector_type(2))) float v2f;
typedef __attribute__((ext_vector_type(8))) float v8f;

// ---------------------------------------------------------------- utilities
__device__ __forceinline__ unsigned hash_u32(unsigned x) {
    x ^= x >> 16; x *= 0x7feb352dU;
    x ^= x >> 15; x *= 0x846ca68bU;
    x ^= x >> 16;
    return x;
}

// ------------------------------------------------- 1) per-neuron mean + EMA
__global__ void mean_ema_kernel(const float* __restrict__ spikes,
                                const float* __restrict__ act_in,
                                float* __restrict__ mean_out,
                                float* __restrict__ ema_out, int N) {
    int i = blockIdx.x * blockDim.x + threadIdx.x;
    if (i >= N) return;
    float s = 0.f;
    #pragma unroll 8
    for (int b = 0; b < NB; ++b) s += spikes[b * N + i];
    float mn = s * (1.0f / NB);
    mean_out[i] = mn;
    ema_out[i]  = 0.99f * act_in[i] + 0.01f * mn;
}

// ------------------------------------------------- 2) column sums of mask
__global__ void colsum_kernel(const float* __restrict__ mask,
                              float* __restrict__ colsum) {
    int j = blockIdx.x * blockDim.x + threadIdx.x;
    if (j >= NPOST) return;
    float s = 0.f;
    #pragma unroll 8
    for (int i = 0; i < NPRE; ++i) s += mask[(size_t)i * NPOST + j];
    colsum[j] = s;
}

// ---------------- 3) streaming pass: hist copy + slot write + mean(HIST)
// Two adjacent (i,j) pairs per thread: 100 contiguous f32 = 25 float4,
// 400B chunks are 16B aligned (even pair * 200B), so loads/stores go b128.
// NPOST is even => both pairs share row i, j1 = j0+1 (never wraps).
// Single read + single write of the 839MB tensor; avg_corr fused in.
__global__ void hist_kernel(const float* __restrict__ hist_in,
                            const float* __restrict__ pre_mean,
                            const float* __restrict__ post_mean,
                            const int*   __restrict__ corr_index_p,
                            float* __restrict__ hist_out,
                            float* __restrict__ avg_corr) {
    const size_t npair2 = (size_t)NPRE * NPOST / 2;
    size_t tid = (size_t)blockIdx.x * blockDim.x + threadIdx.x;
    if (tid >= npair2) return;
    const size_t pair0 = tid * 2;
    const int i  = (int)(pair0 / NPOST);
    const int j0 = (int)(pair0 % NPOST);
    const int cidx = ((corr_index_p[0] % NHIST) + NHIST) % NHIST;
    const float pm = pre_mean[i];
    const float cur0 = pm * post_mean[j0];
    const float cur1 = pm * post_mean[j0 + 1];

    const float4* __restrict__ src = (const float4*)(hist_in + pair0 * NHIST);
    float4*       __restrict__ dst = (float4*)(hist_out + pair0 * NHIST);
    float s0 = 0.f, s1 = 0.f;
    #pragma unroll
    for (int t = 0; t < (2 * NHIST) / 4; ++t) {   // 25 float4 = 100 floats
        float4 v = src[t];
        float* c = &v.x;
        #pragma unroll
        for (int e = 0; e < 4; ++e) {
            const int g = 4 * t + e;              // compile-time constant
            if (g < NHIST) {
                if (g == cidx) c[e] = cur0;
                s0 += c[e];
            } else {
                if (g - NHIST == cidx) c[e] = cur1;
                s1 += c[e];
            }
        }
        dst[t] = v;
    }
    float2 avg;
    avg.x = s0 * (1.0f / NHIST);
    avg.y = s1 * (1.0f / NHIST);
    *(float2*)(avg_corr + pair0) = avg;
}

// ------------------------------------------------- 4) plasticity update
__global__ void update_kernel(const float* __restrict__ mask_in,
                              const float* __restrict__ w_in,
                              const float* __restrict__ avg_corr,
                              const float* __restrict__ colsum,
                              const int*   __restrict__ current_time_p,
                              float* __restrict__ mask_out,
                              float* __restrict__ w_out) {
    size_t idx = (size_t)blockIdx.x * blockDim.x + threadIdx.x;
    if (idx >= (size_t)NPRE * NPOST) return;
    int j = (int)(idx % NPOST);
    float m = mask_in[idx];
    float w = w_in[idx];
    if (current_time_p[0] > 1) {
        // --- formation (hash-RNG stand-in for jax threefry; deterministic)
        bool cand = (avg_corr[idx] > CORR_THR) && (m == 0.0f);
        unsigned h1 = hash_u32((unsigned)idx * 2654435761u ^ 0xDEADBEEFu);
        float u = (float)h1 * 2.3283064365386963e-10f;
        bool col_ok = colsum[j] < MAX_SYN;
        if (cand && (u < FORM_RATE) && col_ok) {
            unsigned h2 = hash_u32(h1 ^ 0x9E3779B9u);
            unsigned h3 = hash_u32(h2 + 0x85EBCA6Bu);
            float u1 = ((float)h2 + 1.0f) * 2.3283064365386963e-10f;
            float u2 = (float)h3 * 2.3283064365386963e-10f;
            float n  = __builtin_sqrtf(-2.0f * __logf(u1)) *
                       __cosf(6.28318530717958647692f * u2);
            m = 1.0f;
            w = 0.1f * n;
        }
        // --- pruning
        if ((__builtin_fabsf(w) < PRUNE_THR) && (m == 1.0f)) {
            m = 0.0f;
            w = 0.0f;
        }
    }
    mask_out[idx] = m;
    w_out[idx]    = w;
}

// ------------- 5) synaptic_current = pre_spikes @ (W * mask) via f32 WMMA
// One wave32 per 16-wide N-tile, holding ALL four 16x16 M-tiles of C
// (M=64) in registers: each fused (W*mask) B tile is loaded exactly once
// and feeds 4 back-to-back WMMAs => W+mask stream from HBM exactly once
// (32MB, the roofline minimum). 128 waves = 16 blocks x 8 wave32s, no
// divergence, EXEC all-ones (WMMA requirement).
// A (16x4 f32): lanes 0-15 hold K={k0,k0+1}, lanes 16-31 hold K={k0+2,k0+3}.
// B (4x16 f32): VGPR0 = rows {K=k0 | K=k0+2}, VGPR1 = rows {K=k0+1 | K=k0+3}.
// C (16x16 f32): VGPR r = rows {M=r | M=r+8}, N = lane%16.
__global__ void __launch_bounds__(256)
gemm_wmma_f32_kernel(const float* __restrict__ A,      // [NB, NPRE]
                     const float* __restrict__ W,      // [NPRE, NPOST]
                     const float* __restrict__ Mk,     // [NPRE, NPOST]
                     float* __restrict__ C) {          // [NB, NPOST]
    const int wave = (blockIdx.x * blockDim.x + threadIdx.x) >> 5; // 0..127
    const int lane = threadIdx.x & 31;
    const int half = lane >> 4;   // 0: lanes 0-15, 1: lanes 16-31
    const int l    = lane & 15;
    const int n0   = wave * 16;   // N-tile origin

    v8f acc0 = {}, acc1 = {}, acc2 = {}, acc3 = {};
    #pragma unroll 2
    for (int k0 = 0; k0 < NPRE; k0 += 4) {
        const int ka = k0 + 2 * half;
        // B tile (shared across all 4 M-tiles), mask fused on load
        const size_t b0 = (size_t)ka * NPOST + n0 + l;
        const size_t b1 = b0 + NPOST;
        v2f b;
        b.x = W[b0] * Mk[b0];
        b.y = W[b1] * Mk[b1];
        // A tiles for M-tiles 0..3 (row = 16*t + l); ka even => 8B aligned
        const float* Arow = A + (size_t)l * NPRE + ka;
        const v2f a0 = *(const v2f*)(Arow);
        const v2f a1 = *(const v2f*)(Arow + 16 * NPRE);
        const v2f a2 = *(const v2f*)(Arow + 32 * NPRE);
        const v2f a3 = *(const v2f*)(Arow + 48 * NPRE);

        acc0 = __builtin_amdgcn_wmma_f32_16x16x4_f32(
            false, a0, false, b, (short)0, acc0, false, false);
        acc1 = __builtin_amdgcn_wmma_f32_16x16x4_f32(
            false, a1, false, b, (short)0, acc1, false, false);
        acc2 = __builtin_amdgcn_wmma_f32_16x16x4_f32(
            false, a2, false, b, (short)0, acc2, false, false);
        acc3 = __builtin_amdgcn_wmma_f32_16x16x4_f32(
            false, a3, false, b, (short)0, acc3, false, false);
    }
    #pragma unroll
    for (int r = 0; r < 8; ++r) {
        const int m = r + 8 * half;
        C[(size_t)(m)      * NPOST + n0 + l] = acc0[r];
        C[(size_t)(m + 16) * NPOST + n0 + l] = acc1[r];
        C[(size_t)(m + 32) * NPOST + n0 + l] = acc2[r];
        C[(size_t)(m + 48) * NPOST + n0 + l] = acc3[r];
    }
}

// --------------------------------------------------------------- dispatch
extern "C" void kernel_launch(void* const* d_in, const int* in_sizes, int n_in,
                              void* d_out, int out_size, void* d_ws, size_t ws_size,
                              hipStream_t stream) {
    const float* pre_spikes   = (const float*)d_in[0];   // [64, 2048]
    const float* post_spikes  = (const float*)d_in[1];   // [64, 2048]
    const float* pre_act      = (const float*)d_in[2];   // [2048]
    const float* post_act     = (const float*)d_in[3];   // [2048]
    const float* hist_in      = (const float*)d_in[4];   // [2048,2048,50]
    const float* mask_in      = (const float*)d_in[5];   // [2048,2048]
    const float* w_in         = (const float*)d_in[6];   // [2048,2048]
    const int*   corr_index   = (const int*)d_in[7];     // scalar
    const int*   current_time = (const int*)d_in[8];     // scalar

    float* out = (float*)d_out;
    // flat output offsets in return order
    const size_t SZ_SC   = (size_t)NB * NPOST;            // 131072
    const size_t SZ_MAT  = (size_t)NPRE * NPOST;          // 4194304
    const size_t SZ_HIST = SZ_MAT * NHIST;                // 209715200
    float* out_sc       = out;                            // synaptic_current
    float* out_mask     = out_sc + SZ_SC;
    float* out_w        = out_mask + SZ_MAT;
    float* out_pre_act  = out_w + SZ_MAT;
    float* out_post_act = out_pre_act + NPRE;
    float* out_hist     = out_post_act + NPOST;
    float* out_avgcorr  = out_hist + SZ_HIST;

    float* ws        = (float*)d_ws;
    float* ws_premn  = ws;                 // [2048]
    float* ws_postmn = ws + NPRE;          // [2048]
    float* ws_colsum = ws + NPRE + NPOST;  // [2048]

    // 1) EMA traces + spike means
    mean_ema_kernel<<<(NPRE + 255) / 256, 256, 0, stream>>>(
        pre_spikes, pre_act, ws_premn, out_pre_act, NPRE);
    mean_ema_kernel<<<(NPOST + 255) / 256, 256, 0, stream>>>(
        post_spikes, post_act, ws_postmn, out_post_act, NPOST);

    // 2) column sums of input mask (formation cap uses pre-update mask)
    colsum_kernel<<<(NPOST + 255) / 256, 256, 0, stream>>>(mask_in, ws_colsum);

    // 3) fused history copy + slot replacement + HIST-mean (839MB stream,
    //    b128 loads/stores, 2 pairs per thread)
    {
        const size_t nthreads = (size_t)NPRE * NPOST / 2;   // 2,097,152
        hist_kernel<<<(unsigned)((nthreads + 255) / 256), 256, 0, stream>>>(
            hist_in, ws_premn, ws_postmn, corr_index, out_hist, out_avgcorr);
    }

    // 4) plasticity: formation + pruning -> new mask / W
    {
        const size_t n = SZ_MAT;
        update_kernel<<<(unsigned)((n + 255) / 256), 256, 0, stream>>>(
            mask_in, w_in, out_avgcorr, ws_colsum, current_time,
            out_mask, out_w);
    }

    // 5) WMMA GEMM on pre-update weights: 128 N-tile waves, 4 M-tiles each
    gemm_wmma_f32_kernel<<<16, 256, 0, stream>>>(
        pre_spikes, w_in, mask_in, out_sc);
}